// SRMBlock_10299331576318
// MI455X (gfx1250) — compile-verified
//
#include <hip/hip_runtime.h>

// SRM block fused kernel for MI455X (gfx1250, wave32).
// out[b,c,h,w] = restore( window_fc( rearrange(x) ) )  -- fully fused, fp32,
// one pass over HBM (memory-bound: ~0.54 GB min traffic, 2.1 GFLOP).

typedef __attribute__((ext_vector_type(2))) float v2f;
typedef __attribute__((ext_vector_type(8))) float v8f;

#define B_ 2
#define C_ 128
#define H_ 512
#define W_ 512

// Rearrangement index map (size=512, M=4, S=2), closed form.
__device__ __forceinline__ int rearr_idx(int k) {
  int i = k >> 2;
  int j = k & 3;
  int p = (i << 2) + j + ((j & 2) ? 4 : 0);   // groups pick padded chunks 4i and 4i+6
  int r = (p < 2) ? p : (p - 2);              // pidx: 2 left edge copies
  r = (p >= 514) ? (p - 4) : r;               // pidx: 2 right edge copies
  return r;
}

// Inverse of the restoration pair-permutation: output pair inv(v) gets xp pair v.
// inv(0)=0, inv(255)=255, even v -> v-1, odd v -> v+1.  Branchless (cndmask).
__device__ __forceinline__ int inv_pair(int v) {
  int t = (v & 1) ? (v + 1) : (v - 1);
  t = (v == 0) ? 0 : t;
  t = (v == 255) ? 255 : t;
  return t;
}

extern "C" __global__ __launch_bounds__(256)
void srm_fused_kernel(const float* __restrict__ x,
                      const float* __restrict__ fc_w,
                      const float* __restrict__ fc_b,
                      float* __restrict__ out) {
  // LDS tile: 16 channels x 4 gathered rows x 32 gathered cols, padded 32->34
  // floats per row so wave-strided v2f reads spread across banks.
  __shared__ float tile[16 * 4 * 34];

  int bid = blockIdx.x;
  const int wwb = bid & 15;  bid >>= 4;   // group of 8 windows along W
  const int wh  = bid & 127; bid >>= 7;   // window row
  const int cb  = bid & 7;   bid >>= 3;   // 16-channel block
  const int b   = bid;

  const int c0  = cb << 4;
  const int ww0 = wwb << 3;
  const size_t HW = (size_t)H_ * W_;

  // ---------------- Stage 1: gathered global -> LDS (lanes along W) --------
  const int tid   = threadIdx.x;
  const int w_idx = tid & 31;            // gathered column 0..31 (8 windows * 4)
  const int q     = tid >> 5;            // 0..7
  const int col   = rearr_idx((ww0 << 2) + w_idx);

#pragma unroll
  for (int r = 0; r < 8; ++r) {
    const int cfh = r * 8 + q;           // 0..63 -> (channel, fh)
    const int c   = cfh >> 2;
    const int fh  = cfh & 3;
    const int row = rearr_idx((wh << 2) + fh);
    const float* gp = x + ((size_t)(b * C_ + c0 + c) * H_ + row) * W_ + col;
    float* lp = &tile[(c * 4 + fh) * 34 + w_idx];
#if __has_builtin(__builtin_amdgcn_global_load_async_to_lds_b32)
    __builtin_amdgcn_global_load_async_to_lds_b32(
        (__attribute__((address_space(1))) int*)gp,
        (__attribute__((address_space(3))) int*)lp, 0, 0);
#else
    *lp = *gp;
#endif
  }
#if __has_builtin(__builtin_amdgcn_global_load_async_to_lds_b32)
#if __has_builtin(__builtin_amdgcn_s_wait_asynccnt)
  __builtin_amdgcn_s_wait_asynccnt(0);
#else
  asm volatile("s_wait_asynccnt 0" ::: "memory");
#endif
#endif
  __syncthreads();

  // ---------------- Stage 2: one 16x16x16 GEMM per wave via 4x WMMA --------
  const int lane = tid & 31;
  const int wave = tid >> 5;             // window ww0 + wave
  const int oc   = lane & 15;            // A row (channel) == B col (o) == D col
  const int hf   = lane >> 4;            // lane half selects K pair / M half
  const int ww   = ww0 + wave;

  // B = fc_w^T fragments: B_k[kk,o] = fc_w[o, 4k+kk]
  //   lane<16: v0=K0,v1=K1 ; lane>=16: v0=K2,v1=K3  (mirrors A layout)
  v2f bfrag[4];
#pragma unroll
  for (int k = 0; k < 4; ++k)
    bfrag[k] = *(const v2f*)(fc_w + oc * 16 + (k << 2) + (hf << 1));
  const float bias = fc_b[oc];

  // A fragments from LDS: A_k[m,kk] = X[c0+m, f = 4k+kk], m = oc
  v2f afrag[4];
#pragma unroll
  for (int k = 0; k < 4; ++k)
    afrag[k] = *(const v2f*)&tile[(oc * 4 + k) * 34 + (wave << 2) + (hf << 1)];

  v8f acc = {};
#pragma unroll
  for (int k = 0; k < 4; ++k)
    acc = __builtin_amdgcn_wmma_f32_16x16x4_f32(
        /*neg_a=*/false, afrag[k], /*neg_b=*/false, bfrag[k],
        /*c_mod=*/(short)0, acc, /*reuse_a=*/false, /*reuse_b=*/false);

  // ---------------- Stage 3: bias + fused restoration scatter --------------
  // D[r + 8*hf, oc]: this lane owns window element o = oc for 8 channels.
  const int hp = oc >> 2;                // row in window
  const int wp = oc & 3;                 // col in window
  const int oh = (inv_pair((wh << 1) + (hp >> 1)) << 1) + (hp & 1);
  const int ow = (inv_pair((ww << 1) + (wp >> 1)) << 1) + (wp & 1);
  float* op = out + ((size_t)(b * C_ + c0 + (hf << 3)) * H_ + oh) * W_ + ow;
#pragma unroll
  for (int r = 0; r < 8; ++r)
    op[(size_t)r * HW] = acc[r] + bias;  // channel stride = 1 MiB, fits imm24
}

extern "C" void kernel_launch(void* const* d_in, const int* in_sizes, int n_in,
                              void* d_out, int out_size, void* d_ws, size_t ws_size,
                              hipStream_t stream) {
  (void)in_sizes; (void)n_in; (void)out_size; (void)d_ws; (void)ws_size;
  const float* x    = (const float*)d_in[0];
  const float* fc_w = (const float*)d_in[1];
  const float* fc_b = (const float*)d_in[2];
  float* out = (float*)d_out;

  // blocks: b(2) * cblk(8) * wh(128) * wwblk(16) = 32768, 256 thr = 8 waves
  const int nblocks = B_ * (C_ / 16) * (H_ / 4) * ((W_ / 4) / 8);
  srm_fused_kernel<<<nblocks, 256, 0, stream>>>(x, fc_w, fc_b, out);
}